// XSimGCL_15994458210395
// MI455X (gfx1250) — compile-verified
//
#include <hip/hip_runtime.h>
#include <cstdint>

#define N_USERS 100000
#define N_ITEMS 50000
#define EMB     64
#define N_NODES (N_USERS + N_ITEMS)
#define NNZ_TOTAL 3000000

constexpr int CHUNK = 256;   // edges staged per chunk (1 per thread)
constexpr int CPB   = 4;     // chunks per block

// ---- CDNA5 async global->LDS staging -------------------------------------
// Per-lane: copies 4 bytes from the global address gptr to the LDS address lds_off; tracked with ASYNCcnt (in-order done).
__device__ __forceinline__ void async_ld_b32(uint32_t lds_off, const void* gptr) {
  asm volatile("global_load_async_to_lds_b32 %0, %1, off"
               :: "v"(lds_off), "v"(gptr)
               : "memory");
}
__device__ __forceinline__ void wait_async0() {
  asm volatile("s_wait_asynccnt 0x0" ::: "memory");
}
__device__ __forceinline__ void wait_async3() {
  asm volatile("s_wait_asynccnt 0x3" ::: "memory");
}

// Non-returning f32 atomic add: guaranteed native global_atomic_add_f32
// (no CAS loop, no VGPR writeback; tracked on STOREcnt, flushed at S_ENDPGM).
__device__ __forceinline__ void atomic_add_f32(float* p, float v) {
  asm volatile("global_atomic_add_f32 %0, %1, off"
               :: "v"(p), "v"(v)
               : "memory");
}

// ---- SpMM: xnext[r,:] += v * xprev[c,:]  (COO, atomic scatter) ------------
// 256 threads = 8 waves. Edge triples double-buffered into LDS via the
// CDNA5 async-to-LDS path as 16B packed records {row, col, val, pad};
// each wave processes 32 edges/chunk, lane l owns dims {2l,2l+1}:
// one ds_load_b128 (edge record) + one b64 gather + two global f32 atomics.
__global__ __launch_bounds__(256) void xsimgcl_spmm(
    const int*   __restrict__ rows, const int* __restrict__ cols,
    const float* __restrict__ vals,
    const float* __restrict__ xprev, float* __restrict__ xnext) {
  __shared__ int sE[2][CHUNK][4];   // [buf][edge][{row,col,val,pad}] : 8 KB

  const int  t      = threadIdx.x;
  const int  lane   = t & 31;
  const int  wfirst = t & ~31;                       // wave's first thread id
  const long blockBase = (long)blockIdx.x * (CPB * CHUNK);

  // prologue: stage chunk 0 into buffer 0
  {
    long e = blockBase + t;
    if (e < NNZ_TOTAL) {
      async_ld_b32((uint32_t)(uintptr_t)&sE[0][t][0], rows + e);
      async_ld_b32((uint32_t)(uintptr_t)&sE[0][t][1], cols + e);
      async_ld_b32((uint32_t)(uintptr_t)&sE[0][t][2], vals + e);
    }
  }

  for (int k = 0; k < CPB; ++k) {
    const int cur = k & 1;

    // stage chunk k+1 into the other buffer while chunk k is still in flight
    bool didnext = false;
    if (k + 1 < CPB) {
      const int  nb    = (k + 1) & 1;
      const long nbase = blockBase + (long)(k + 1) * CHUNK;
      didnext = (nbase + wfirst) < NNZ_TOTAL;        // wave-uniform
      long e = nbase + t;
      if (e < NNZ_TOTAL) {
        async_ld_b32((uint32_t)(uintptr_t)&sE[nb][t][0], rows + e);
        async_ld_b32((uint32_t)(uintptr_t)&sE[nb][t][1], cols + e);
        async_ld_b32((uint32_t)(uintptr_t)&sE[nb][t][2], vals + e);
      }
    }
    // async loads complete in order: <=3 outstanding means chunk k is in LDS
    if (didnext) wait_async3(); else wait_async0();
    __syncthreads();

    const long cbase = blockBase + (long)k * CHUNK;
    int cnt  = (int)((long)NNZ_TOTAL - cbase < (long)CHUNK
                         ? (long)NNZ_TOTAL - cbase : (long)CHUNK);
    int jBeg = (t >> 5) * 32;                         // this wave's 32 edges
    int jEnd = jBeg + 32 < cnt ? jBeg + 32 : cnt;

#pragma unroll 4
    for (int j = jBeg; j < jEnd; ++j) {
      const int4 ed = *(const int4*)&sE[cur][j][0];   // uniform -> ds_load_b128
      const int   r = ed.x;
      const int   c = ed.y;
      const float v = __int_as_float(ed.z);
      const float2 xv = *(const float2*)(xprev + ((size_t)c << 6) + (lane << 1));
      float* dst = xnext + ((size_t)r << 6) + (lane << 1);
      atomic_add_f32(dst,     v * xv.x);
      atomic_add_f32(dst + 1, v * xv.y);
    }
    __syncthreads();   // buffer 'cur' may be overwritten next iteration
  }
}

// ---- init: x0 = concat(user_emb, item_emb); acc(out) = x0; zero first xn --
__global__ __launch_bounds__(256) void xsimgcl_init(
    const float* __restrict__ ue, const float* __restrict__ ie,
    float* __restrict__ xprev, float* __restrict__ out,
    float* __restrict__ xzero) {
  const size_t total = (size_t)N_NODES * (EMB / 4);
  size_t i = (size_t)blockIdx.x * 256 + threadIdx.x;   // float4 index
  if (i >= total) return;
  size_t node = i >> 4;          // /(EMB/4)
  size_t q    = i & 15;
  const float4 v = (node < N_USERS)
      ? ((const float4*)(ue + node * EMB))[q]
      : ((const float4*)(ie + (node - N_USERS) * EMB))[q];
  ((float4*)xprev)[i] = v;
  ((float4*)out)[i]   = v;
  ((float4*)xzero)[i] = make_float4(0.f, 0.f, 0.f, 0.f);
}

// ---- acc: out = (out + x) * scale; optionally zero next layer's target ----
__global__ __launch_bounds__(256) void xsimgcl_acc(
    float* __restrict__ out, const float* __restrict__ x,
    float* __restrict__ xzero, float scale) {
  const size_t total = (size_t)N_NODES * (EMB / 4);
  size_t i = (size_t)blockIdx.x * 256 + threadIdx.x;
  if (i >= total) return;
  float4 o = ((float4*)out)[i];
  float4 v = ((const float4*)x)[i];
  o.x = (o.x + v.x) * scale;
  o.y = (o.y + v.y) * scale;
  o.z = (o.z + v.z) * scale;
  o.w = (o.w + v.w) * scale;
  ((float4*)out)[i] = o;
  if (xzero) ((float4*)xzero)[i] = make_float4(0.f, 0.f, 0.f, 0.f);
}

extern "C" void kernel_launch(void* const* d_in, const int* in_sizes, int n_in,
                              void* d_out, int out_size, void* d_ws, size_t ws_size,
                              hipStream_t stream) {
  const float* ue   = (const float*)d_in[0];
  const float* ie   = (const float*)d_in[1];
  const int*   rows = (const int*)d_in[2];
  const int*   cols = (const int*)d_in[3];
  const float* vals = (const float*)d_in[4];
  float*       out  = (float*)d_out;

  const size_t nodeFloats = (size_t)N_NODES * EMB;   // 9.6M floats
  float* xA = (float*)d_ws;                           // 38.4 MB
  float* xB = xA + nodeFloats;                        // 38.4 MB

  const size_t totalV4 = nodeFloats / 4;
  const unsigned gridEl = (unsigned)((totalV4 + 255) / 256);

  // xA = x0, out = x0, xB = 0 (ready as first spmm target)
  xsimgcl_init<<<gridEl, 256, 0, stream>>>(ue, ie, xA, out, xB);

  const int nChunks = (NNZ_TOTAL + CHUNK - 1) / CHUNK;
  const int nBlocks = (nChunks + CPB - 1) / CPB;

  float* xp = xA;
  float* xn = xB;
  for (int layer = 0; layer < 3; ++layer) {
    xsimgcl_spmm<<<nBlocks, 256, 0, stream>>>(rows, cols, vals, xp, xn);
    const float scale = (layer == 2) ? 0.25f : 1.0f;
    // old xp is dead after this spmm -> zero it here so it can be the next xn
    float* zbuf = (layer == 2) ? nullptr : xp;
    xsimgcl_acc<<<gridEl, 256, 0, stream>>>(out, xn, zbuf, scale);
    float* tmp = xp; xp = xn; xn = tmp;
  }
}